// FarthestPointSample_63256278335593
// MI455X (gfx1250) — compile-verified
//
#include <hip/hip_runtime.h>
#include <stdint.h>

// Farthest Point Sampling, MI455X (gfx1250, wave32).
// B=16 batches, N=131072 points, D=3, 1024 centroids.
//
// Design: state fully register-resident. 32 blocks per batch x 256 threads;
// each thread owns 16 points (x,y,z,dist in VGPRs). Per round: 16 local
// distance updates, wave32 shuffle-reduce of a packed u64 argmax key,
// ds_max_u64 per wave into LDS, one global_atomic_max_u64 + arrival counter
// per block into L2, spin until all 32 blocks of the batch arrived, then
// broadcast the winning point. Round-tagged keys are monotonically increasing,
// so atomicMax slots never need resetting (parity double-buffer + counter
// ordering makes reuse race-free).
//
// CDNA5 path: initial coordinate tile (3 x 4096 f32, strided by N) is DMA'd
// into LDS by the Tensor Data Mover (tensor_load_to_lds + s_wait_tensorcnt).

#define BATCHES          16
#define NPTS             131072
#define NCENT            1024
#define BLOCKS_PER_BATCH 32
#define THREADS          256
#define PTS_PER_BLOCK    (NPTS / BLOCKS_PER_BATCH)   // 4096
#define PTS_PER_THREAD   (PTS_PER_BLOCK / THREADS)   // 16

typedef unsigned int u32;
typedef unsigned long long u64;

typedef unsigned int uint32x4 __attribute__((ext_vector_type(4)));
typedef int          int32x8  __attribute__((ext_vector_type(8)));
typedef int          int32x4  __attribute__((ext_vector_type(4)));

// ws layout: u64 slots[BATCHES][2] (parity-double-buffered argmax keys),
// then u32 counts[BATCHES] (monotonic arrival counters). 320 bytes total.
__global__ __launch_bounds__(64)
void fps_init_ws(u64* __restrict__ slots, u32* __restrict__ counts) {
  int t = threadIdx.x;
  if (t < BATCHES * 2) slots[t] = 0ull;
  if (t < BATCHES)     counts[t] = 0u;
}

__global__ __launch_bounds__(THREADS)
void fps_kernel(const float* __restrict__ pt, int* __restrict__ out,
                u64* __restrict__ slots, u32* __restrict__ counts) {
  __shared__ float ldsC[3 * PTS_PER_BLOCK];   // 48 KB staged coordinate tile
  __shared__ u64   ldsKey;                    // block-level argmax (monotonic)
  __shared__ float ldsP[3];                   // broadcast centroid coords
  __shared__ u32   ldsWin;                    // broadcast winner index

  const int b    = blockIdx.x / BLOCKS_PER_BATCH;
  const int bib  = blockIdx.x % BLOCKS_PER_BATCH;
  const int tid  = threadIdx.x;
  const u32 base = (u32)bib * PTS_PER_BLOCK;
  const float* batchPt = pt + (size_t)b * 3u * NPTS;

  if (tid == 0) {
    ldsKey  = 0ull;
    ldsP[0] = batchPt[0];            // initial centroid = point 0
    ldsP[1] = batchPt[NPTS];
    ldsP[2] = batchPt[2u * NPTS];
  }

  // ---- Stage this block's [3 x 4096] tile (row stride N) into LDS ----
#if __has_builtin(__builtin_amdgcn_tensor_load_to_lds)
  if (tid < 32) {  // one wave issues the TDM op (EXEC ignored by TDM)
    unsigned long long ga = (unsigned long long)(uintptr_t)(batchPt + base);
    u32 ldsOff = (u32)(uintptr_t)(void*)&ldsC[0];  // low 32 bits = LDS offset
    uint32x4 g0;
    g0[0] = 0x1u;                                   // count=1, user descriptor
    g0[1] = ldsOff;                                 // lds_addr
    g0[2] = (u32)ga;                                // global_addr[31:0]
    g0[3] = ((u32)(ga >> 32) & 0x01FFFFFFu) | 0x80000000u; // addr[56:32]|type=2
    int32x8 g1;
    g1[0] = 0x00020000;                             // data_size=4B, no flags
    g1[1] = (int)(((u32)NPTS & 0xFFFFu) << 16);     // tensor_dim0[15:0]
    g1[2] = (int)(((u32)NPTS >> 16) | (3u << 16));  // tensor_dim0 hi | dim1 lo
    g1[3] = (int)((u32)PTS_PER_BLOCK << 16);        // dim1 hi(0) | tile_dim0
    g1[4] = 3;                                      // tile_dim1=3, tile_dim2=0
    g1[5] = (int)NPTS;                              // tensor_dim0_stride lo
    g1[6] = 0;                                      // stride hi / dim1_stride lo
    g1[7] = 0;
    int32x4 gz  = {0, 0, 0, 0};                     // <=2D: groups 2/3 unused
    int32x8 g4z = {0, 0, 0, 0, 0, 0, 0, 0};         // group 4 unused
    __builtin_amdgcn_tensor_load_to_lds(g0, g1, gz, gz, g4z, 0);
    __builtin_amdgcn_s_wait_tensorcnt((short)0);
  }
#else
  for (int i = tid; i < 3 * PTS_PER_BLOCK; i += THREADS) {
    int row = i >> 12;                  // / 4096
    int col = i & (PTS_PER_BLOCK - 1);  // % 4096
    ldsC[i] = batchPt[(size_t)row * NPTS + base + col];
  }
#endif
  __syncthreads();

  // ---- Unpack tile into registers; dist starts at +inf ----
  float X[PTS_PER_THREAD], Y[PTS_PER_THREAD], Z[PTS_PER_THREAD], Dm[PTS_PER_THREAD];
#pragma unroll
  for (int k = 0; k < PTS_PER_THREAD; ++k) {
    int l = k * THREADS + tid;          // ascending k == ascending global idx
    X[k]  = ldsC[l];
    Y[k]  = ldsC[PTS_PER_BLOCK + l];
    Z[k]  = ldsC[2 * PTS_PER_BLOCK + l];
    Dm[k] = __builtin_inff();
  }

  float px = ldsP[0], py = ldsP[1], pz = ldsP[2];
  u32 cur = 0;  // meaningful on (bib==0, tid==0) only

  for (int it = 0; it < NCENT; ++it) {
    if (bib == 0 && tid == 0) out[b * NCENT + it] = (int)cur;

    // Local distance update + argmax over this thread's 16 points.
    float bD = -1.0f; int bK = 0;
#pragma unroll
    for (int k = 0; k < PTS_PER_THREAD; ++k) {
      float dx = __fadd_rn(X[k], -px);
      float dy = __fadd_rn(Y[k], -py);
      float dz = __fadd_rn(Z[k], -pz);
      // left-assoc, uncontracted: matches (dx*dx + dy*dy) + dz*dz
      float d = __fadd_rn(__fadd_rn(__fmul_rn(dx, dx), __fmul_rn(dy, dy)),
                          __fmul_rn(dz, dz));
      float m = fminf(Dm[k], d);
      Dm[k] = m;
      if (m > bD) { bD = m; bK = k; }   // strict > keeps lowest index on ties
    }
    u32 gidx = base + (u32)bK * THREADS + (u32)tid;   // idx within batch
    // Monotonic packed key: round tag | dist bits | inverted index.
    u64 key = ((u64)(u32)(it + 1) << 49) |
              ((u64)__float_as_uint(bD) << 17) |
              (u64)(NPTS - 1u - gidx);

    // wave32 shuffle-reduce
#pragma unroll
    for (int off = 16; off > 0; off >>= 1) {
      u64 o = __shfl_xor(key, off, 32);
      key = (o > key) ? o : key;
    }
    if ((tid & 31) == 0) atomicMax(&ldsKey, key);     // ds_max_u64, 8 waves
    __syncthreads();

    if (tid == 0) {
      u64 blkKey = ldsKey;  // monotonic; stale lower-tag values are dominated
      u64* slot = &slots[b * 2 + (it & 1)];
      __hip_atomic_fetch_max(slot, blkKey, __ATOMIC_RELAXED,
                             __HIP_MEMORY_SCOPE_AGENT);
      __hip_atomic_fetch_add(&counts[b], 1u, __ATOMIC_RELEASE,
                             __HIP_MEMORY_SCOPE_AGENT);
      const u32 target = (u32)(it + 1) * BLOCKS_PER_BATCH;
      while (__hip_atomic_load(&counts[b], __ATOMIC_ACQUIRE,
                               __HIP_MEMORY_SCOPE_AGENT) < target)
        __builtin_amdgcn_s_sleep(1);
      u64 w = __hip_atomic_load(slot, __ATOMIC_RELAXED,
                                __HIP_MEMORY_SCOPE_AGENT);
      u32 win = NPTS - 1u - (u32)(w & 0x1FFFFu);
      ldsWin  = win;
      ldsP[0] = batchPt[win];
      ldsP[1] = batchPt[NPTS + win];
      ldsP[2] = batchPt[2u * NPTS + win];
    }
    __syncthreads();
    px = ldsP[0]; py = ldsP[1]; pz = ldsP[2];
    if (tid == 0) cur = ldsWin;
  }
}

extern "C" void kernel_launch(void* const* d_in, const int* in_sizes, int n_in,
                              void* d_out, int out_size, void* d_ws, size_t ws_size,
                              hipStream_t stream) {
  (void)in_sizes; (void)n_in; (void)out_size; (void)ws_size;
  const float* pt = (const float*)d_in[0];   // [16, 3, 131072] f32
  // d_in[1] = num_centroids (==1024), compile-time constant here.
  int* out = (int*)d_out;                    // [16, 1024] int32 indices

  u64* slots  = (u64*)d_ws;                  // [16][2]
  u32* counts = (u32*)(slots + BATCHES * 2); // [16]

  fps_init_ws<<<1, 64, 0, stream>>>(slots, counts);
  fps_kernel<<<dim3(BATCHES * BLOCKS_PER_BATCH), dim3(THREADS), 0, stream>>>(
      pt, out, slots, counts);
}